// ResizeToSameSize_21698174779469
// MI455X (gfx1250) — compile-verified
//
#include <hip/hip_runtime.h>
#include <hip/hip_bf16.h>
#include <stdint.h>

// ResizeToSameSize: concat of 12 channels at 352x352 fp32.
// Channel 0 = copy; channels 1..11 = nearest-neighbor block upsample with
// power-of-two scales (2,4,8,16,32 repeating). Pure bandwidth kernel:
// ~0.95 MB read + ~5.95 MB write. One float4 of output per thread.
//
// CDNA5 path: for upsample channels, each block stages its <=3 needed source
// rows (<=528 floats) into LDS with GLOBAL_LOAD_ASYNC_TO_LDS_B32 (ASYNCcnt),
// then serves the 2-32x redundant re-reads from LDS broadcast.

#define TARGET 352
#define QUADS_PER_ROW 88            // 352 / 4
#define QUADS_PER_CHAN 30976        // 352*352/4
#define BLOCKS_PER_CHAN 121         // 30976 / 256
#define SMEM_FLOATS 544             // >= 3 rows * 176 floats = 528

struct SrcPtrs {
    const float* p[12];
};

__global__ __launch_bounds__(256)
void ResizeToSameSize_21698174779469_kernel(SrcPtrs srcs, float* __restrict__ out) {
    __shared__ float smem[SMEM_FLOATS];

    const int c = blockIdx.y;                                   // channel: block-uniform (SALU)
    const int tid = threadIdx.x;
    const int q = blockIdx.x * 256 + tid;                       // float4 index in channel [0,30976)
    const int y = q / QUADS_PER_ROW;                            // output row (const-divisor -> mulhi)
    const int xq = q - y * QUADS_PER_ROW;                       // float4 column index [0,88)

    const float* __restrict__ src = srcs.p[c];
    float4 v;

    if (c == 0) {
        // Straight streaming copy, 128-bit loads + gfx1250 prefetch stream.
        const float4* __restrict__ src4 = (const float4*)src;
        __builtin_prefetch(src4 + q + 512, 0, 0);
        v = src4[q];
    } else {
        // shift = ((c-1) % 5) + 1  (scales 2,4,8,16,32 for src 176,88,44,22,11)
        const int shift = ((c - 1) % 5) + 1;                    // block-uniform SALU math
        const int ssize = TARGET >> shift;                      // source width/height

        // Block-level source-row window (pure blockIdx math -> scalar).
        const int q0 = blockIdx.x * 256;
        const int y0 = q0 / QUADS_PER_ROW;
        const int y1 = (q0 + 255) / QUADS_PER_ROW;
        const int sy0 = y0 >> shift;
        const int sy1 = y1 >> shift;
        const int nfl = (sy1 - sy0 + 1) * ssize;                // floats to stage (<= 528)

        // Async DMA the contiguous source-row window into LDS (ASYNCcnt path).
        const float* __restrict__ gsrc = src + sy0 * ssize;
        for (int i = tid; i < nfl; i += 256) {
            uint64_t gaddr = (uint64_t)(uintptr_t)(gsrc + i);
            uint32_t laddr = (uint32_t)(uintptr_t)(&smem[i]);
            asm volatile("global_load_async_to_lds_b32 %0, %1, off"
                         :: "v"(laddr), "v"(gaddr) : "memory");
        }
        asm volatile("s_wait_asynccnt 0" ::: "memory");          // my wave's DMAs landed in LDS
        __syncthreads();                                         // all waves' DMAs visible

        const int lrow = (y >> shift) - sy0;                     // row within staged window
        const float* __restrict__ srow = &smem[lrow * ssize];
        if (shift == 1) {
            // 4 output cols -> 2 source cols: {a,a,b,b}  (LDS float2 read)
            const float2 t = ((const float2*)srow)[xq];
            v = make_float4(t.x, t.x, t.y, t.y);
        } else {
            // 4 output cols -> 1 source col: LDS broadcast read
            const float a = srow[xq >> (shift - 2)];
            v = make_float4(a, a, a, a);
        }
    }

    // Coalesced 128-bit store: wave32 x 16B = 512B contiguous per wave.
    ((float4* __restrict__)out)[c * QUADS_PER_CHAN + q] = v;
}

extern "C" void kernel_launch(void* const* d_in, const int* in_sizes, int n_in,
                              void* d_out, int out_size, void* d_ws, size_t ws_size,
                              hipStream_t stream) {
    (void)in_sizes; (void)n_in; (void)out_size; (void)d_ws; (void)ws_size;

    SrcPtrs srcs;
    for (int i = 0; i < 12; ++i) {
        srcs.p[i] = (const float*)d_in[i];
    }
    float* out = (float*)d_out;

    dim3 grid(BLOCKS_PER_CHAN, 12, 1);   // 121 blocks/channel x 12 channels
    dim3 block(256, 1, 1);               // 8 waves (wave32)
    ResizeToSameSize_21698174779469_kernel<<<grid, block, 0, stream>>>(srcs, out);
}